// MeshConv_6940667150714
// MI455X (gfx1250) — compile-verified
//
#include <hip/hip_runtime.h>

// MeshConv: out[e,:] = [x[e], min(n0,n1), max(n0,n1), min(n2,n3), max(n2,n3)] @ W^T + b
// fp32 WMMA (V_WMMA_F32_16X16X4_F32) path for gfx1250 — exact fp32 numerics.

typedef float v2f __attribute__((ext_vector_type(2)));
typedef float v8f __attribute__((ext_vector_type(8)));

#define K5 320          // 5*C
#define NSTEPS 80       // K5 / 4
#define BLOCK_ROWS 128  // 8 waves * 16 rows

// Permuted-K -> original-K mapping.
// kp in [0,64):    x channels (identity)
// kp in [64,192):  pair1, chunks of 8 = {min01[4j..4j+3], max01[4j..4j+3]}
// kp in [192,320): pair2, same interleave
__device__ __forceinline__ int kperm_to_korig(int kp) {
  if (kp < 64) return kp;
  int t = kp - 64;
  int seg = t >> 7;   // 0 = pair1, 1 = pair2
  t &= 127;
  int j = t >> 3;     // 4-channel chunk
  int r = t & 7;      // 0..3 -> min, 4..7 -> max
  return 64 + seg * 128 + ((r & 4) << 4) + 4 * j + (r & 3);
}

__global__ __launch_bounds__(256) void meshconv_wmma_f32(
    const float* __restrict__ x, const int* __restrict__ nbr,
    const float* __restrict__ W, const float* __restrict__ bias,
    float* __restrict__ out, int E)
{
  // B fragments, pre-swizzled for wave32 WMMA: [s][nt][lane][v], v2f per (lane,nt,s).
  // ldsB[((s*4+nt)*32 + lane)*2 + v] = W[nt*16+m][korig(4s + 2h + v)], lane = h*16+m.
  __shared__ float ldsB[NSTEPS * 4 * 32 * 2];   // 80 KB
  __shared__ float ldsBias[64];

  const int tid = threadIdx.x;

  // ---- Stage permuted W^T as WMMA B-fragments (one v2f per slot) ----
  for (int i = tid; i < NSTEPS * 4 * 32; i += 256) {
    int s   = i >> 7;       // / (4*32)
    int rem = i & 127;
    int nt  = rem >> 5;
    int ln  = rem & 31;
    int hh  = ln >> 4;
    int mm  = ln & 15;
    int n   = nt * 16 + mm;
    int k0  = kperm_to_korig(4 * s + 2 * hh);   // korig(kp+1) == k0+1 by construction
    ((v2f*)ldsB)[i] = *(const v2f*)(W + n * K5 + k0);
  }
  if (tid < 64) ldsBias[tid] = bias[tid];
  __syncthreads();

  const int lane = tid & 31;
  const int wave = tid >> 5;
  const int m    = lane & 15;
  const int h    = lane >> 4;
  const int off  = 2 * h;   // K sub-offset for this half of the wave

  const long rowBase = (long)blockIdx.x * BLOCK_ROWS + wave * 16;
  long eA = rowBase + m;
  if (eA >= E) eA = E - 1;  // clamp: WMMA requires EXEC all-1s; stores are guarded

  const float* px = x + eA * 64;
  int4 nb = ((const int4*)nbr)[eA];
  const float* p0 = x + (long)(nb.x < 0 ? 0 : nb.x) * 64;
  const float* p1 = x + (long)(nb.y < 0 ? 0 : nb.y) * 64;
  const float* p2 = x + (long)(nb.z < 0 ? 0 : nb.z) * 64;
  const float* p3 = x + (long)(nb.w < 0 ? 0 : nb.w) * 64;
  const bool z0 = nb.x < 0, z1 = nb.y < 0, z2 = nb.z < 0, z3 = nb.w < 0;

  v8f acc0 = {}, acc1 = {}, acc2 = {}, acc3 = {};
  const v2f* ldsBv = (const v2f*)ldsB;

#define WMMA_STEP(S, AV)                                                        \
  do {                                                                          \
    const v2f* bp = ldsBv + (S) * 4 * 32 + lane;                                \
    acc0 = __builtin_amdgcn_wmma_f32_16x16x4_f32(false, (AV), false, bp[0],     \
                                                 (short)0, acc0, false, false); \
    acc1 = __builtin_amdgcn_wmma_f32_16x16x4_f32(false, (AV), false, bp[32],    \
                                                 (short)0, acc1, false, false); \
    acc2 = __builtin_amdgcn_wmma_f32_16x16x4_f32(false, (AV), false, bp[64],    \
                                                 (short)0, acc2, false, false); \
    acc3 = __builtin_amdgcn_wmma_f32_16x16x4_f32(false, (AV), false, bp[96],    \
                                                 (short)0, acc3, false, false); \
  } while (0)

  // ---- Segment 0: self features x (steps 0..15) ----
  for (int j = 0; j < 16; ++j) {
    v2f a = *(const v2f*)(px + 4 * j + off);
    WMMA_STEP(j, a);
  }

  // ---- Segment 1: sorted pair (n0, n1) (steps 16..47, min/max interleaved) ----
  for (int j = 0; j < 16; ++j) {
    v2f va = *(const v2f*)(p0 + 4 * j + off);
    v2f vb = *(const v2f*)(p1 + 4 * j + off);
    va.x = z0 ? 0.0f : va.x;  va.y = z0 ? 0.0f : va.y;
    vb.x = z1 ? 0.0f : vb.x;  vb.y = z1 ? 0.0f : vb.y;
    v2f amin = { fminf(va.x, vb.x), fminf(va.y, vb.y) };
    v2f amax = { fmaxf(va.x, vb.x), fmaxf(va.y, vb.y) };
    WMMA_STEP(16 + 2 * j, amin);
    WMMA_STEP(17 + 2 * j, amax);
  }

  // ---- Segment 2: sorted pair (n2, n3) (steps 48..79) ----
  for (int j = 0; j < 16; ++j) {
    v2f va = *(const v2f*)(p2 + 4 * j + off);
    v2f vb = *(const v2f*)(p3 + 4 * j + off);
    va.x = z2 ? 0.0f : va.x;  va.y = z2 ? 0.0f : va.y;
    vb.x = z3 ? 0.0f : vb.x;  vb.y = z3 ? 0.0f : vb.y;
    v2f amin = { fminf(va.x, vb.x), fminf(va.y, vb.y) };
    v2f amax = { fmaxf(va.x, vb.x), fmaxf(va.y, vb.y) };
    WMMA_STEP(48 + 2 * j, amin);
    WMMA_STEP(49 + 2 * j, amax);
  }
#undef WMMA_STEP

  // ---- Bias + store (C/D layout: VGPR r -> row r + 8h, lane -> col nt*16+m) ----
  const float b0 = ldsBias[m];
  const float b1 = ldsBias[16 + m];
  const float b2 = ldsBias[32 + m];
  const float b3 = ldsBias[48 + m];
  for (int r = 0; r < 8; ++r) {
    long row = rowBase + r + 8 * h;
    if (row < E) {
      float* po = out + row * 64 + m;
      po[0]  = acc0[r] + b0;
      po[16] = acc1[r] + b1;
      po[32] = acc2[r] + b2;
      po[48] = acc3[r] + b3;
    }
  }
}

extern "C" void kernel_launch(void* const* d_in, const int* in_sizes, int n_in,
                              void* d_out, int out_size, void* d_ws, size_t ws_size,
                              hipStream_t stream) {
  const float* x   = (const float*)d_in[0];
  const int*   nbr = (const int*)d_in[1];
  const float* W   = (const float*)d_in[2];
  const float* b   = (const float*)d_in[3];
  float*       out = (float*)d_out;

  const int E = in_sizes[0] / 64;
  const int blocks = (E + BLOCK_ROWS - 1) / BLOCK_ROWS;
  meshconv_wmma_f32<<<blocks, 256, 0, stream>>>(x, nbr, W, b, out, E);
}